// PennyLaneQuantumLayer_78374563217582
// MI455X (gfx1250) — compile-verified
//
#include <hip/hip_runtime.h>
#include <math.h>

// ---------------------------------------------------------------------------
// 10-qubit, depth-6 PennyLane-style variational circuit, batch 8192.
// Fully fused single kernel; real-valued statevector in LDS; gate passes are
// 16x16 x (16x64) GEMMs driven by v_wmma_f32_16x16x4_f32.
// Pass geometry is specialized at compile time so every LDS address is a
// single affine op (base + g*stride), no runtime branching.
// ---------------------------------------------------------------------------

typedef float v2f __attribute__((ext_vector_type(2)));
typedef float v8f __attribute__((ext_vector_type(8)));

#define NQ     10
#define DIM    1024
#define DEPTH  6
#define WAVES  8          // samples per workgroup (1 wave = 1 sample)
#define TPB    (WAVES * 32)

// RY(theta) = [[c,-s],[s,c]]   entry (i,j)
__device__ __forceinline__ float ryE(float c, float s, int i, int j) {
    return (i == j) ? c : (j ? -s : s);
}

// Apply one fused 16x16 gate pass:  S <- U(16x16) @ S(16x64), in place.
//   PASS 0: gate dim = wires 0-3 (MSB nibble)  idx = g*64 + j        stride 64
//   PASS 1: gate dim = wires 4-7               idx = a*64 + g*4 + c  stride 4
//           column j = a0*32 + ah*4 + c, a = (ah<<1)|a0; a0 = q3 selects the
//           matrix variant with the folded CNOT(3,4); a0 == t>>1 (constant
//           per unrolled tile).
//   PASS 2: gate dim = wires 6-9 (LSB nibble)  idx = j*16 + g        stride 1
template <int PASS>
__device__ __forceinline__ void applyPass(float* __restrict__ Sw,
                                          const float (&U)[4][16][16],
                                          int l16, int hi) {
#pragma unroll
    for (int t = 0; t < 4; ++t) {
        const int um = (PASS == 0) ? 0 : (PASS == 1 ? 1 + (t >> 1) : 3);
        const int j  = t * 16 + l16;           // spectator column for this lane

        int base, stride;
        if (PASS == 0) {
            base = j;                  stride = 64;
        } else if (PASS == 1) {
            const int c  = j & 3;
            const int ah = (j >> 2) & 7;
            const int a0 = t >> 1;             // == j>>5, compile-time per tile
            base = (((ah << 1) | a0) << 6) + c; stride = 4;
        } else {
            base = j << 4;             stride = 1;
        }

        v2f A[4], B[4];
#pragma unroll
        for (int k = 0; k < 4; ++k) {
            // A 16x4 f32 layout: lane row = l16; VGPR0 <-> K=k0, VGPR1 <-> K=k0+1
            const int k0 = k * 4 + 2 * hi;
            A[k].x = U[um][l16][k0];
            A[k].y = U[um][l16][k0 + 1];
            // B 4x16 f32 layout mirrors A: lane col = l16, rows k0 / k0+1
            B[k].x = Sw[base + k0 * stride];
            B[k].y = Sw[base + (k0 + 1) * stride];
        }
        v8f acc = {0.f, 0.f, 0.f, 0.f, 0.f, 0.f, 0.f, 0.f};
#pragma unroll
        for (int k = 0; k < 4; ++k)
            acc = __builtin_amdgcn_wmma_f32_16x16x4_f32(
                      false, A[k], false, B[k], (short)0, acc, false, false);
        // C/D 16x16 f32 layout: VGPR v, lane -> row = v + 8*hi, col = l16
#pragma unroll
        for (int v = 0; v < 8; ++v)
            Sw[base + (v + 8 * hi) * stride] = acc[v];
    }
}

__global__ __launch_bounds__(TPB)
void qsim_kernel(const float* __restrict__ x,      // (B, 10)
                 const float* __restrict__ w,      // (6, 10)
                 float* __restrict__ out)          // (B, 10)
{
    __shared__ float S[WAVES][DIM];   // one 1024-float state slab per wave
    __shared__ float U[4][16][16];    // layer gate matrices (batch-invariant)

    const int tid    = threadIdx.x;
    const int lane   = tid & 31;
    const int wid    = tid >> 5;
    const int sample = blockIdx.x * WAVES + wid;
    float* Sw = S[wid];

    // ---- embedding: post-(H, RY(x*pi/2)) state is a product state ----------
    {
        float v0[NQ], v1[NQ];
        const float isq2 = 0.70710678118654752440f;
#pragma unroll
        for (int q = 0; q < NQ; ++q) {
            float h = x[sample * NQ + q] * 0.78539816339744830962f; // (x*pi/2)/2
            float sh, ch;
            __sincosf(h, &sh, &ch);
            v0[q] = (ch - sh) * isq2;   // (RY*H)|0> component 0
            v1[q] = (ch + sh) * isq2;   // (RY*H)|0> component 1
        }
#pragma unroll 4
        for (int r = 0; r < 32; ++r) {
            const int i = r * 32 + lane;
            float amp = 1.0f;
#pragma unroll
            for (int q = 0; q < NQ; ++q)
                amp *= ((i >> (9 - q)) & 1) ? v1[q] : v0[q];
            Sw[i] = amp;
        }
    }

    const int l16 = lane & 15;
    const int hi  = lane >> 4;    // 0 or 1 (half-wave select in WMMA layouts)

    // ---- 6 variational layers ----------------------------------------------
    for (int l = 0; l < DEPTH; ++l) {
        __syncthreads();   // all waves done with previous layer's U matrices

        // Build the 4 fused 16x16 gate matrices (256 threads, 4 entries each).
        // Left-multiplied CNOT permutations become a branchless row remap.
        {
            float c[NQ], s[NQ];
#pragma unroll
            for (int q = 0; q < NQ; ++q) {
                float h = 0.5f * w[l * NQ + q];
                __sincosf(h, &s[q], &c[q]);
            }
            const int row = tid >> 4, col = tid & 15;

            // sigma chain for CNOT(b3->b2),(b2->b1),(b1->b0) inside a nibble:
            // apply innermost first: if bit1 flip bit0; if bit2 flip bit1; ...
            int r = row;
            r ^= (r >> 1) & 1;            // sigma(2,3)
            r ^= ((r >> 2) & 1) << 1;     // sigma(1,2)
            r ^= ((r >> 3) & 1) << 2;     // sigma(0,1)

            // pass 0: wires 0-3
            U[0][row][col] = ryE(c[0], s[0], (r >> 3) & 1, (col >> 3) & 1)
                           * ryE(c[1], s[1], (r >> 2) & 1, (col >> 2) & 1)
                           * ryE(c[2], s[2], (r >> 1) & 1, (col >> 1) & 1)
                           * ryE(c[3], s[3],  r       & 1,  col       & 1);

            // pass 1: wires 4-7 (same sigma pattern); two variants folding
            // CNOT(3,4) as X on wire4 when spectator bit q3 (= a0) is 1.
            const float f5 = ryE(c[5], s[5], (r >> 2) & 1, (col >> 2) & 1);
            const float f6 = ryE(c[6], s[6], (r >> 1) & 1, (col >> 1) & 1);
            const float f7 = ryE(c[7], s[7],  r       & 1,  col       & 1);
            const float tail = f5 * f6 * f7;
#pragma unroll
            for (int a0 = 0; a0 < 2; ++a0)
                U[1 + a0][row][col] =
                    ryE(c[4], s[4], ((r >> 3) & 1) ^ a0, (col >> 3) & 1) * tail;

            // pass 2: wires 6-9; identity on bits 3,2; RY8,RY9 + C78,C89
            int r3 = row;
            r3 ^= (r3 >> 1) & 1;          // sigma(8,9)
            r3 ^= ((r3 >> 2) & 1) << 1;   // sigma(7,8)
            float v = 0.0f;
            if ((r3 >> 2) == (col >> 2))
                v = ryE(c[8], s[8], (r3 >> 1) & 1, (col >> 1) & 1)
                  * ryE(c[9], s[9],  r3       & 1,  col       & 1);
            U[3][row][col] = v;
        }
        __syncthreads();

        applyPass<0>(Sw, U, l16, hi);
        applyPass<1>(Sw, U, l16, hi);
        applyPass<2>(Sw, U, l16, hi);
    }

    // ---- expectations: <Z_w> = sum_i sign_w(i) * |s_i|^2 -------------------
    float acc[NQ];
#pragma unroll
    for (int q = 0; q < NQ; ++q) acc[q] = 0.0f;
#pragma unroll 4
    for (int r = 0; r < 32; ++r) {
        const int i = r * 32 + lane;
        const float a = Sw[i];
        const float p = a * a;
#pragma unroll
        for (int q = 0; q < NQ; ++q)
            acc[q] += ((i >> (9 - q)) & 1) ? -p : p;
    }
#pragma unroll
    for (int off = 16; off > 0; off >>= 1)
#pragma unroll
        for (int q = 0; q < NQ; ++q)
            acc[q] += __shfl_xor(acc[q], off, 32);
    if (lane == 0)
#pragma unroll
        for (int q = 0; q < NQ; ++q)
            out[sample * NQ + q] = acc[q];
}

extern "C" void kernel_launch(void* const* d_in, const int* in_sizes, int n_in,
                              void* d_out, int out_size, void* d_ws, size_t ws_size,
                              hipStream_t stream) {
    (void)n_in; (void)out_size; (void)d_ws; (void)ws_size;
    const float* x = (const float*)d_in[0];   // (8192, 10) float32
    const float* w = (const float*)d_in[1];   // (6, 10)    float32
    float* out = (float*)d_out;               // (8192, 10) float32

    const int batch  = in_sizes[0] / NQ;      // 8192
    const int blocks = batch / WAVES;         // 1024 workgroups, 8 waves each
    qsim_kernel<<<blocks, TPB, 0, stream>>>(x, w, out);
}